// customLSTM_10325101380317
// MI455X (gfx1250) — compile-verified
//
#include <hip/hip_runtime.h>
#include <hip/hip_bf16.h>

// ---------------- problem constants ----------------
#define T_STEPS 365
#define BATCH   512
#define IDIM    32
#define HDIM    256
#define GDIM    1024            // 4*H
#define KDIM    288             // H + I (concatenated K for recurrent + input proj)
#define KSTEPS  72              // KDIM / 4
#define KPAIRS  36              // two k-steps per b128 fragment load
#define NTILES  64              // GDIM / 16
#define WAVES   8
#define NT_PER_WAVE 8           // 64 tiles / 8 waves
#define BTILE   16

typedef __attribute__((ext_vector_type(2))) float v2f;
typedef __attribute__((ext_vector_type(4))) float v4f;
typedef __attribute__((ext_vector_type(8))) float v8f;

// ---------------------------------------------------------------------------
// Pack [w_hh | w_ih] (K' = 256+32 = 288) into WMMA f32 16x16x4 B-fragment
// order: wfrag[((ntile*KPAIRS + j)*32 + lane)*4 + q]
//   q0,q1 -> k-step s0=2j   : B[4*s0 + 2*hi + {0,1}][n0 + l15]
//   q2,q3 -> k-step s1=2j+1 : B[4*s1 + 2*hi + {0,1}][n0 + l15]
// where B[k][n] = (k < 256) ? w_hh[n][k] : w_ih[n][k-256]   (B = W^T)
// ---------------------------------------------------------------------------
extern "C" __global__ void lstm_pack_weights(const float* __restrict__ w_hh,
                                             const float* __restrict__ w_ih,
                                             float* __restrict__ wfrag) {
    int g = blockIdx.x * blockDim.x + threadIdx.x;
    const int total = NTILES * KPAIRS * 32;
    if (g >= total) return;
    int lane  = g & 31;
    int j     = (g >> 5) % KPAIRS;
    int ntile = g / (KPAIRS * 32);
    int l15 = lane & 15;
    int hi  = lane >> 4;
    int n   = ntile * 16 + l15;

    v4f v;
    #pragma unroll
    for (int q = 0; q < 4; ++q) {
        int s = 2 * j + (q >> 1);
        int k = 4 * s + 2 * hi + (q & 1);
        float val = (k < HDIM) ? w_hh[(size_t)n * HDIM + k]
                               : w_ih[(size_t)n * IDIM + (k - HDIM)];
        v[q] = val;
    }
    *(v4f*)(wfrag + ((size_t)g << 2)) = v;
}

// ---------------------------------------------------------------------------
// Persistent LSTM kernel: one workgroup owns a 16-row batch tile for all 365
// steps. 8 waves; wave w owns gate columns [128w, 128w+128) = 8 WMMA N-tiles.
// Dynamic LDS layout (floats):
//   hx   [16][288]   : cols 0..255 = h_t, cols 256..287 = x_t      (4608)
//   gl   [16][1024]  : pre-activation gates                        (16384)
//   bias [1024]      : b_ih + b_hh                                 (1024)
//   wl   [256]       : w_lin                                       (256)
// ---------------------------------------------------------------------------
extern "C" __global__ void
__launch_bounds__(256)
lstm_persistent(const float* __restrict__ x,
                const float* __restrict__ wfrag,
                const float* __restrict__ b_ih,
                const float* __restrict__ b_hh,
                const float* __restrict__ w_lin,
                const float* __restrict__ b_lin,
                float* __restrict__ out) {
    extern __shared__ float smem[];
    float* hx   = smem;                       // 16*288
    float* gl   = hx + BTILE * KDIM;          // 16*1024
    float* bias = gl + BTILE * GDIM;          // 1024
    float* wl   = bias + GDIM;                // 256

    const int tid  = threadIdx.x;             // 0..255
    const int wave = tid >> 5;
    const int lane = tid & 31;
    const int l15  = lane & 15;
    const int hi   = lane >> 4;
    const int b0   = blockIdx.x * BTILE;

    // ---- one-time init ----
    for (int i = tid; i < GDIM; i += 256) bias[i] = b_ih[i] + b_hh[i];
    if (tid < HDIM) wl[tid] = w_lin[tid];
    for (int i = tid; i < BTILE * KDIM; i += 256) hx[i] = 0.0f;   // h0 = 0
    const float blin = b_lin[0];
    __syncthreads();   // init complete before first x staging (different threads
                       // touch the x-region of hx in the staging loop below)

    float c_reg[BTILE];
    #pragma unroll
    for (int r = 0; r < BTILE; ++r) c_reg[r] = 0.0f;              // c0 = 0

    const size_t S = (size_t)BATCH * T_STEPS * HDIM;
    float* hs_o   = out;
    float* cs_o   = out + S;
    float* gate_o = out + 2 * S;              // i,f,g,o each S apart
    float* y_o    = out + 6 * S;

    // per-wave fragment pointers
    const float* wfw  = wfrag + ((size_t)(wave * NT_PER_WAVE) * KPAIRS * 32) * 4;
    const float* arow = hx + l15 * KDIM + 2 * hi;

    for (int t = 0; t < T_STEPS; ++t) {
        // ---- stage x_t tile into hx columns 256..287 ----
        #pragma unroll
        for (int i2 = 0; i2 < 2; ++i2) {
            int e = tid + 256 * i2;           // 0..511
            int m  = e >> 5;
            int ii = e & 31;
            hx[m * KDIM + HDIM + ii] =
                x[((size_t)t * BATCH + (b0 + m)) * IDIM + ii];
        }
        __syncthreads();   // x staged; prior-step h writes visible

        // ---- GEMM: [16 x K'=288] @ [K' x 128 slice] via WMMA f32 16x16x4 ----
        v8f acc[NT_PER_WAVE];
        #pragma unroll
        for (int i = 0; i < NT_PER_WAVE; ++i)
            acc[i] = (v8f){0.f, 0.f, 0.f, 0.f, 0.f, 0.f, 0.f, 0.f};

        for (int j = 0; j < KPAIRS; ++j) {
            v4f bfr[NT_PER_WAVE];
            #pragma unroll
            for (int i = 0; i < NT_PER_WAVE; ++i)
                bfr[i] = *(const v4f*)(wfw +
                    ((((size_t)i * KPAIRS + j) * 32 + lane) << 2));

            v2f a0 = *(const v2f*)(arow + 8 * j);        // k-step 2j
            v2f a1 = *(const v2f*)(arow + 8 * j + 4);    // k-step 2j+1

            #pragma unroll
            for (int i = 0; i < NT_PER_WAVE; ++i) {
                v2f bv = {bfr[i].x, bfr[i].y};
                acc[i] = __builtin_amdgcn_wmma_f32_16x16x4_f32(
                    false, a0, false, bv, (short)0, acc[i], false, false);
            }
            #pragma unroll
            for (int i = 0; i < NT_PER_WAVE; ++i) {
                v2f bv = {bfr[i].z, bfr[i].w};
                acc[i] = __builtin_amdgcn_wmma_f32_16x16x4_f32(
                    false, a1, false, bv, (short)0, acc[i], false, false);
            }
        }

        // ---- spill gates (+bias) to LDS: C layout lane l, vgpr r ->
        //      row m = r + 8*hi, col n = ntile*16 + l15 ----
        #pragma unroll
        for (int i = 0; i < NT_PER_WAVE; ++i) {
            int n = (wave * NT_PER_WAVE + i) * 16 + l15;
            float bn = bias[n];
            #pragma unroll
            for (int r = 0; r < 8; ++r)
                gl[(r + 8 * hi) * GDIM + n] = acc[i][r] + bn;
        }
        __syncthreads();   // gates complete

        // ---- elementwise LSTM cell: thread owns hidden column `tid` ----
        {
            const int col = tid;  // 0..255
            #pragma unroll
            for (int r = 0; r < BTILE; ++r) {
                float iv = gl[r * GDIM + col];
                float fv = gl[r * GDIM + HDIM + col];
                float gv = gl[r * GDIM + 2 * HDIM + col];
                float ov = gl[r * GDIM + 3 * HDIM + col];

                float si = 1.0f / (1.0f + __expf(-iv));
                float sf = 1.0f / (1.0f + __expf(-fv));
                float so = 1.0f / (1.0f + __expf(-ov));
                float tg = tanhf(gv);

                float c1 = c_reg[r] * sf + si * tg;
                float h1 = so * tanhf(c1);
                c_reg[r] = c1;
                hx[r * KDIM + col] = h1;                  // h for next step

                size_t base = ((size_t)(b0 + r) * T_STEPS + t) * HDIM + col;
                hs_o[base]            = h1;
                cs_o[base]            = c1;
                gate_o[base]          = iv;   // raw pre-activation gates
                gate_o[S + base]      = fv;
                gate_o[2 * S + base]  = gv;
                gate_o[3 * S + base]  = ov;
            }
        }
        __syncthreads();   // h_t+1 complete

        // ---- y_hat = h1 . w_lin + b_lin : wave w reduces rows 2w, 2w+1 ----
        #pragma unroll
        for (int rr = 0; rr < 2; ++rr) {
            int r = wave * 2 + rr;
            float s = 0.0f;
            for (int idx = lane; idx < HDIM; idx += 32)
                s += hx[r * KDIM + idx] * wl[idx];
            #pragma unroll
            for (int off = 16; off > 0; off >>= 1)
                s += __shfl_down(s, off, 32);
            if (lane == 0)
                y_o[(size_t)(b0 + r) * T_STEPS + t] = s + blin;
        }
        // next iteration's x staging touches only the x-region; safe.
    }
}

// ---------------------------------------------------------------------------
extern "C" void kernel_launch(void* const* d_in, const int* in_sizes, int n_in,
                              void* d_out, int out_size, void* d_ws, size_t ws_size,
                              hipStream_t stream) {
    const float* x     = (const float*)d_in[0];
    const float* w_ih  = (const float*)d_in[1];
    const float* w_hh  = (const float*)d_in[2];
    const float* b_ih  = (const float*)d_in[3];
    const float* b_hh  = (const float*)d_in[4];
    const float* w_lin = (const float*)d_in[5];
    const float* b_lin = (const float*)d_in[6];
    float* out   = (float*)d_out;
    float* wfrag = (float*)d_ws;   // 64*36*32*4 floats = ~1.13 MB

    const int packN = NTILES * KPAIRS * 32;
    lstm_pack_weights<<<(packN + 255) / 256, 256, 0, stream>>>(w_hh, w_ih, wfrag);

    const size_t shm = (size_t)(BTILE * KDIM + BTILE * GDIM + GDIM + HDIM) * sizeof(float);
    lstm_persistent<<<BATCH / BTILE, WAVES * 32, shm, stream>>>(
        x, wfrag, b_ih, b_hh, w_lin, b_lin, out);
}